// PP_14491219657328
// MI455X (gfx1250) — compile-verified
//
#include <hip/hip_runtime.h>

// GCN (3 conv layers, cached gcn_norm) for MI455X / gfx1250.
// Dense GEMMs: V_WMMA_F32_16X16X4_F32, W staged in padded LDS (ds_load_b64),
// A fragments software-pipelined one K-step ahead.
// Graph scatter: wave-per-edge global_atomic_add_f32 (L2-resident, 192MB L2).

namespace {

constexpr int NN  = 100000;   // nodes
constexpr int NE  = 1600000;  // edges
constexpr int IND = 256;      // input dim
constexpr int H   = 64;       // hidden dim

typedef __attribute__((ext_vector_type(2))) float v2f;
typedef __attribute__((ext_vector_type(8))) float v8f;

__global__ __launch_bounds__(256) void k_fill(float* __restrict__ p, float v, int n) {
  int i = blockIdx.x * 256 + threadIdx.x;
  if (i < n) p[i] = v;
}

// deg[col[e]] += w[e]   (deg pre-filled with 1.0 for the self-loop)
__global__ __launch_bounds__(256) void k_deg(const int* __restrict__ col,
                                             const float* __restrict__ w,
                                             float* __restrict__ deg, int e) {
  int i = blockIdx.x * 256 + threadIdx.x;
  if (i < e) atomicAdd(&deg[col[i]], w[i]);
}

__global__ __launch_bounds__(256) void k_rsqrt(float* __restrict__ d, int n) {
  int i = blockIdx.x * 256 + threadIdx.x;
  if (i < n) d[i] = rsqrtf(d[i]);
}

// norm[e] = dis[row[e]] * w[e] * dis[col[e]]
__global__ __launch_bounds__(256) void k_norm(const int* __restrict__ row,
                                              const int* __restrict__ col,
                                              const float* __restrict__ w,
                                              const float* __restrict__ dis,
                                              float* __restrict__ nrm, int e) {
  int i = blockIdx.x * 256 + threadIdx.x;
  if (i < e) nrm[i] = dis[row[i]] * w[i] * dis[col[i]];
}

// C[M x 64] = A[M x K] * W[64 x K]^T using V_WMMA_F32_16X16X4_F32.
// W is staged in LDS (row stride K+4 -> conflict-free ds_load_b64 across the
// 16-lane halves). One wave owns a 16(M) x 64(N) tile; 8 waves / block.
// A fragment: lane m=lane&15 holds A[m][k+kb..k+kb+1], kb=(lane>>4)*2.
// C/D: VGPR r, lanes 0-15 -> M=r, lanes 16-31 -> M=8+r, N=lane&15.
template <int K>
__global__ __launch_bounds__(256) void k_gemm_nt64(const float* __restrict__ A,
                                                   const float* __restrict__ W,
                                                   float* __restrict__ C, int M) {
  constexpr int LDW = K + 4;  // padded LDS row stride (spreads banks)
  __shared__ float sW[64 * LDW];

  // Cooperative load of W[64 x K] into padded LDS (K is constexpr pow2).
  for (int idx = threadIdx.x; idx < 64 * K; idx += 256) {
    int r = idx / K;
    int c = idx - r * K;
    sW[r * LDW + c] = W[idx];
  }
  __syncthreads();

  const int lane    = threadIdx.x & 31;
  const int wave    = threadIdx.x >> 5;
  const int rowBase = (blockIdx.x * 8 + wave) * 16;
  if (rowBase >= M) return;  // wave-uniform -> EXEC all-ones for WMMA

  const int m  = lane & 15;
  const int kb = (lane >> 4) << 1;  // 0 or 2
  int arow = rowBase + m;
  if (arow >= M) arow = M - 1;      // clamped rows masked at store time
  const float* Ap = A + (size_t)arow * K + kb;
  const float* Wl = &sW[m * LDW + kb];

  v8f c0 = {}, c1 = {}, c2 = {}, c3 = {};
  v2f aNext = *(const v2f*)(Ap);  // k = 0, in flight early
#pragma unroll 4
  for (int k = 0; k < K; k += 4) {
    v2f a = aNext;
    if (k + 4 < K) aNext = *(const v2f*)(Ap + k + 4);  // prefetch next step
    v2f bb0 = *(const v2f*)(Wl + 0 * 16 * LDW + k);
    v2f bb1 = *(const v2f*)(Wl + 1 * 16 * LDW + k);
    v2f bb2 = *(const v2f*)(Wl + 2 * 16 * LDW + k);
    v2f bb3 = *(const v2f*)(Wl + 3 * 16 * LDW + k);
    c0 = __builtin_amdgcn_wmma_f32_16x16x4_f32(false, a, false, bb0, (short)0, c0, false, false);
    c1 = __builtin_amdgcn_wmma_f32_16x16x4_f32(false, a, false, bb1, (short)0, c1, false, false);
    c2 = __builtin_amdgcn_wmma_f32_16x16x4_f32(false, a, false, bb2, (short)0, c2, false, false);
    c3 = __builtin_amdgcn_wmma_f32_16x16x4_f32(false, a, false, bb3, (short)0, c3, false, false);
  }

  const int mOff = (lane >> 4) << 3;  // 0 or 8
  const int n    = lane & 15;
#pragma unroll
  for (int r = 0; r < 8; ++r) {
    int orow = rowBase + mOff + r;
    if (orow < M) {
      float* Cr = C + (size_t)orow * 64 + n;
      Cr[0]  = c0[r];
      Cr[16] = c1[r];
      Cr[32] = c2[r];
      Cr[48] = c3[r];
    }
  }
}

// acc = t * dis^2 (self-loop term) + bias, float4-vectorized (16 float4 / row)
__global__ __launch_bounds__(256) void k_selfinit(const float4* __restrict__ t,
                                                  const float* __restrict__ dis,
                                                  const float4* __restrict__ b4,
                                                  float4* __restrict__ acc, int total4) {
  int i = blockIdx.x * 256 + threadIdx.x;
  if (i >= total4) return;
  float d = dis[i >> 4];
  d = d * d;
  float4 tv = t[i];
  float4 bv = b4[i & 15];
  float4 o;
  o.x = tv.x * d + bv.x;
  o.y = tv.y * d + bv.y;
  o.z = tv.z * d + bv.z;
  o.w = tv.w * d + bv.w;
  acc[i] = o;
}

// One wave per edge: acc[col[e]][:] += t[row[e]][:] * norm[e]
// Each lane handles 2 contiguous floats (64 floats per row, wave32).
__global__ __launch_bounds__(256) void k_scatter(const int* __restrict__ row,
                                                 const int* __restrict__ col,
                                                 const float* __restrict__ nrm,
                                                 const float* __restrict__ t,
                                                 float* __restrict__ acc, int e) {
  const int lane = threadIdx.x & 31;
  const int edge = blockIdx.x * 8 + (threadIdx.x >> 5);
  if (edge >= e) return;
  const int   r = row[edge];   // wave-uniform
  const int   c = col[edge];   // wave-uniform
  const float w = nrm[edge];
  v2f s = *(const v2f*)(t + (size_t)r * 64 + lane * 2);
  float* dst = acc + (size_t)c * 64 + lane * 2;
  atomicAdd(dst + 0, s.x * w);
  atomicAdd(dst + 1, s.y * w);
}

// In-place ReLU, float4-vectorized
__global__ __launch_bounds__(256) void k_relu(float4* __restrict__ p, int total4) {
  int i = blockIdx.x * 256 + threadIdx.x;
  if (i >= total4) return;
  float4 v = p[i];
  v.x = fmaxf(v.x, 0.0f);
  v.y = fmaxf(v.y, 0.0f);
  v.z = fmaxf(v.z, 0.0f);
  v.w = fmaxf(v.w, 0.0f);
  p[i] = v;
}

}  // namespace

extern "C" void kernel_launch(void* const* d_in, const int* in_sizes, int n_in,
                              void* d_out, int out_size, void* d_ws, size_t ws_size,
                              hipStream_t stream) {
  (void)in_sizes; (void)n_in; (void)out_size; (void)ws_size;

  const float* x    = (const float*)d_in[0];
  const int*   ei   = (const int*)d_in[1];
  const float* ew   = (const float*)d_in[2];
  const float* embW = (const float*)d_in[3];
  const float* W0   = (const float*)d_in[4];
  const float* b0   = (const float*)d_in[5];
  const float* W1   = (const float*)d_in[6];
  const float* b1   = (const float*)d_in[7];
  float*       out  = (float*)d_out;

  const int* erow = ei;        // edge_index[0]
  const int* ecol = ei + NE;   // edge_index[1]

  // Workspace carve-out (256B aligned): dis | norm | hA | hB  (~58 MB)
  char* ws = (char*)d_ws;
  auto alignUp = [](size_t v) { return (v + 255) & ~(size_t)255; };
  size_t off = 0;
  float* dis = (float*)(ws + off); off = alignUp(off + (size_t)NN * 4);
  float* nrm = (float*)(ws + off); off = alignUp(off + (size_t)NE * 4);
  float* hA  = (float*)(ws + off); off = alignUp(off + (size_t)NN * H * 4);
  float* hB  = (float*)(ws + off); off = alignUp(off + (size_t)NN * H * 4);

  const dim3 blk(256);
  const int NH4   = NN * H / 4;
  const int gN    = (NN + 255) / 256;
  const int gE    = (NE + 255) / 256;
  const int gNH4  = (NH4 + 255) / 256;
  const int gGemm = (NN + 127) / 128;  // 8 waves * 16 rows per block
  const int gScat = (NE + 7) / 8;      // 8 edges (waves) per block

  // --- gcn_norm (cached; reused by all three layers) ---
  k_fill <<<gN, blk, 0, stream>>>(dis, 1.0f, NN);  // deg starts at self-loop weight
  k_deg  <<<gE, blk, 0, stream>>>(ecol, ew, dis, NE);
  k_rsqrt<<<gN, blk, 0, stream>>>(dis, NN);
  k_norm <<<gE, blk, 0, stream>>>(erow, ecol, ew, dis, nrm, NE);

  // --- embed: hA = x @ embW^T (K=256) ---
  k_gemm_nt64<IND><<<gGemm, blk, 0, stream>>>(x, embW, hA, NN);

  // --- layer 0: hA = relu(conv(hA, W0, b0)) ---
  k_gemm_nt64<H><<<gGemm, blk, 0, stream>>>(hA, W0, hB, NN);   // hA now dead -> acc
  k_selfinit<<<gNH4,  blk, 0, stream>>>((const float4*)hB, dis, (const float4*)b0,
                                        (float4*)hA, NH4);
  k_scatter <<<gScat, blk, 0, stream>>>(erow, ecol, nrm, hB, hA, NE);
  k_relu    <<<gNH4,  blk, 0, stream>>>((float4*)hA, NH4);

  // --- layer 1: hA = relu(conv(hA, W1, b1)) ---
  k_gemm_nt64<H><<<gGemm, blk, 0, stream>>>(hA, W1, hB, NN);
  k_selfinit<<<gNH4,  blk, 0, stream>>>((const float4*)hB, dis, (const float4*)b1,
                                        (float4*)hA, NH4);
  k_scatter <<<gScat, blk, 0, stream>>>(erow, ecol, nrm, hB, hA, NE);
  k_relu    <<<gNH4,  blk, 0, stream>>>((float4*)hA, NH4);

  // --- final layer (reference bug: reuses W1/b1, no relu) -> d_out directly ---
  k_gemm_nt64<H><<<gGemm, blk, 0, stream>>>(hA, W1, hB, NN);
  k_selfinit<<<gNH4,  blk, 0, stream>>>((const float4*)hB, dis, (const float4*)b1,
                                        (float4*)out, NH4);
  k_scatter <<<gScat, blk, 0, stream>>>(erow, ecol, nrm, hB, out, NE);
}